// QuantumKernel_65481071403967
// MI455X (gfx1250) — compile-verified
//
#include <hip/hip_runtime.h>
#include <hip/hip_bf16.h>
#include <math.h>

typedef __attribute__((ext_vector_type(2))) float v2f;
typedef __attribute__((ext_vector_type(8))) float v8f;

#define N_WIRES   24
#define HALF_BITS 12            // 12 wires per half
#define HALF_DIM  4096          // 2^12
#define TILE      16
#define TILES_PER_SIDE (HALF_DIM / TILE)          // 256
#define WAVES_PER_BLOCK 8
#define TOTAL_TILES (TILES_PER_SIDE * TILES_PER_SIDE) // 65536

// ---------------------------------------------------------------------------
// Kernel 1: build the two half-state factor tables.
//   A[h] = prod_{b=0..11} ( bit(h,b) ? |sin(y[11-b]/2)| : |cos(y[11-b]/2)| )
//   B[l] = prod_{b=0..11} ( bit(l,b) ? |sin(y[23-b]/2)| : |cos(y[23-b]/2)| )
// (wire i selects bit 23-i of the flat index; angles are -y, abs kills signs)
// ---------------------------------------------------------------------------
__global__ void qstate_factors_kernel(const float* __restrict__ y,
                                      float* __restrict__ Atab,
                                      float* __restrict__ Btab) {
    __shared__ float fc[N_WIRES];
    __shared__ float fs[N_WIRES];
    const int tid = threadIdx.x;
    if (tid < N_WIRES) {
        float h = y[tid] * 0.5f;
        fc[tid] = fabsf(cosf(h));   // |cos(-y/2)| == |cos(y/2)|
        fs[tid] = fabsf(sinf(h));   // |sin(-y/2)| == |sin(y/2)|
    }
    __syncthreads();
    for (int k = tid; k < HALF_DIM; k += blockDim.x) {
        float pa = 1.0f, pb = 1.0f;
#pragma unroll
        for (int b = 0; b < HALF_BITS; ++b) {
            const int bit = (k >> b) & 1;
            pa *= bit ? fs[HALF_BITS - 1 - b] : fc[HALF_BITS - 1 - b];
            pb *= bit ? fs[N_WIRES  - 1 - b] : fc[N_WIRES  - 1 - b];
        }
        Atab[k] = pa;
        Btab[k] = pb;
    }
}

// ---------------------------------------------------------------------------
// Kernel 2: out[h*4096 + l] = A[h] * B[l] as a rank-1 WMMA outer product.
// Each wave computes one 16x16 f32 tile via V_WMMA_F32_16X16X4_F32 with only
// the K=0 slice of A (16x4) and B (4x16) populated; all other K slices are
// zeroed in both operands so their products vanish regardless of lane layout.
// Store-bandwidth bound: 64 MB out -> ~2.75 us floor at 23.3 TB/s.
// ---------------------------------------------------------------------------
__global__ void qstate_outer_wmma_kernel(const float* __restrict__ Atab,
                                         const float* __restrict__ Btab,
                                         float* __restrict__ out) {
    const int lane = threadIdx.x & 31;
    const int wave = threadIdx.x >> 5;

    const int t  = blockIdx.x * WAVES_PER_BLOCK + wave;  // tile id, < 65536
    const int tM = t >> 8;    // row-tile index   (A / high index bits)
    const int tN = t & 255;   // col-tile index   (B / low index bits)

    // Branchless lane masking: lanes 0..15 carry the K=0 operand slices
    // (A: M=lane, K=0 ; B: K=0, N=lane). EXEC stays all-ones for the WMMA.
    const float mask = (lane < 16) ? 1.0f : 0.0f;
    const int   li   = lane & 15;

    const float av = Atab[tM * TILE + li] * mask;
    const float bv = Btab[tN * TILE + li] * mask;

    v2f a; a.x = av;  a.y = 0.0f;   // VGPR1 (K=1/3 slice) = 0
    v2f b; b.x = bv;  b.y = 0.0f;
    v8f c = {};

    // D = A(16x4) x B(4x16) + 0  -> rank-1 outer product tile
    c = __builtin_amdgcn_wmma_f32_16x16x4_f32(
            /*neg_a=*/false, a, /*neg_b=*/false, b,
            /*c_mod=*/(short)0, c, /*reuse_a=*/false, /*reuse_b=*/false);

    // C/D layout: VGPR r, lanes 0-15 -> (M=r, N=lane); lanes 16-31 -> (M=8+r).
    const int n     = lane & 15;
    const int mBase = (lane >> 4) << 3;   // 0 or 8
    float* __restrict__ outp =
        out + (((long long)tM * TILE) << HALF_BITS) + (long long)tN * TILE + n;
#pragma unroll
    for (int r = 0; r < 8; ++r) {
        outp[((long long)(mBase + r)) << HALF_BITS] = c[r];
    }
}

// ---------------------------------------------------------------------------
// Host-side launcher
// ---------------------------------------------------------------------------
extern "C" void kernel_launch(void* const* d_in, const int* in_sizes, int n_in,
                              void* d_out, int out_size, void* d_ws, size_t ws_size,
                              hipStream_t stream) {
    // inputs: d_in[0] = x (unused, dead in reference), d_in[1] = y (24 floats)
    const float* y = (const float*)d_in[1];
    float* out  = (float*)d_out;          // 2^24 floats
    float* Atab = (float*)d_ws;           // 4096 floats
    float* Btab = Atab + HALF_DIM;        // 4096 floats (32 KB total scratch)

    qstate_factors_kernel<<<1, 256, 0, stream>>>(y, Atab, Btab);

    const int blocks = TOTAL_TILES / WAVES_PER_BLOCK;  // 8192
    qstate_outer_wmma_kernel<<<blocks, WAVES_PER_BLOCK * 32, 0, stream>>>(
        Atab, Btab, out);
}